// SmallfryEmbedding_87162066305578
// MI455X (gfx1250) — compile-verified
//
#include <hip/hip_runtime.h>

// Smallfry embedding decode for MI455X (gfx1250, wave32).
// Pure gather workload: HBM-bound (~157 MB total traffic => ~6.7us floor at
// 23.3 TB/s). No matmul => WMMA is inapplicable; we use the CDNA5 async
// global->LDS DMA path (ASYNCcnt) to stage the codebooks, LDS for the 19.6M
// codebook gathers, b128 coalesced code loads, and non-temporal b128 stores
// for the write-once output stream.

#define DIMS          300
#define CHUNKS_PER_ROW 75          // 300 / 4
#define NCODES         16
#define NBLOCKS        98
#define CB_FLOATS     (NBLOCKS * NCODES)   // 1568 floats
#define CB_B128       ((CB_FLOATS * 4) / 16) // 392 16-byte chunks
#define BLOCK_SHIFT    12          // BLOCK_SIZE = 4096

typedef __attribute__((ext_vector_type(4))) float float4v;
typedef __attribute__((ext_vector_type(4))) int   int4v;

__global__ __launch_bounds__(256) void
smallfry_decode_kernel(const int* __restrict__ input,
                       const int* __restrict__ codes,
                       const float* __restrict__ codebooks,
                       float* __restrict__ out,
                       int n_chunks, int n_tokens)
{
    __shared__ float lds_cb[CB_FLOATS];   // 6272 B, fits easily in 320KB/WGP LDS

    // ---- Stage codebooks into LDS via CDNA5 async DMA (ASYNCcnt path) ----
    // Each active lane copies 16 bytes from the global base in saddr plus offset in vaddr to the LDS address in vdst.
    {
        const unsigned lds_base = (unsigned)(size_t)(&lds_cb[0]); // low 32b of
        // the flat addrspacecast == LDS byte offset (HW truncates addr[31:0]).
        for (int i = threadIdx.x; i < CB_B128; i += blockDim.x) {
            unsigned lds_off = lds_base + (unsigned)i * 16u;
            unsigned g_off   = (unsigned)i * 16u;
            asm volatile("global_load_async_to_lds_b128 %0, %1, %2"
                         :
                         : "v"(lds_off), "v"(g_off), "s"(codebooks)
                         : "memory");
        }
        asm volatile("s_wait_asynccnt 0x0" ::: "memory");
    }
    __syncthreads();

    // ---- Main gather loop: one 16B chunk of output per thread-iteration ----
    const int stride = (int)(gridDim.x * blockDim.x);
    for (int gidx = (int)(blockIdx.x * blockDim.x + threadIdx.x);
         gidx < n_chunks; gidx += stride) {
        const unsigned token = (unsigned)gidx / CHUNKS_PER_ROW;
        const unsigned j     = (unsigned)gidx - token * CHUNKS_PER_ROW;

        // Prefetch the token-id stream ahead (global_prefetch_b8).
        {
            unsigned pf = token + 2048u;
            if (pf < (unsigned)n_tokens)
                __builtin_prefetch(input + pf, 0, 1);
        }

        const int tok = input[token];                 // shared by 75 lanes -> L0 hit
        const unsigned cb_row = ((unsigned)tok >> BLOCK_SHIFT) * NCODES;

        // Coalesced 128-bit load of 4 codes (row base 1200B => 16B aligned).
        const int4v c4 = *(const int4v*)(codes + (size_t)tok * DIMS + (size_t)j * 4u);

        float4v o;
        o.x = lds_cb[cb_row + ((unsigned)c4.x & 15u)];
        o.y = lds_cb[cb_row + ((unsigned)c4.y & 15u)];
        o.z = lds_cb[cb_row + ((unsigned)c4.z & 15u)];
        o.w = lds_cb[cb_row + ((unsigned)c4.w & 15u)];

        // Output is written once and never re-read: stream it past the caches.
        __builtin_nontemporal_store(o, (float4v*)out + gidx);
    }
}

extern "C" void kernel_launch(void* const* d_in, const int* in_sizes, int n_in,
                              void* d_out, int out_size, void* d_ws, size_t ws_size,
                              hipStream_t stream) {
    const int*   input     = (const int*)d_in[0];    // [32*2048] token ids
    const int*   codes     = (const int*)d_in[1];    // [400000*300] 4-bit codes (as int32)
    const float* codebooks = (const float*)d_in[2];  // [98*16] f32
    float*       out       = (float*)d_out;          // [32*2048*300] f32

    const int n_tokens = in_sizes[0];
    const int n_chunks = n_tokens * CHUNKS_PER_ROW;  // 16B output chunks

    const int threads = 256;                          // 8 waves (wave32)
    int blocks = (n_chunks + threads - 1) / threads;
    if (blocks > 4096) blocks = 4096;                 // grid-stride; amortize LDS staging
    if (blocks < 1)    blocks = 1;

    smallfry_decode_kernel<<<blocks, threads, 0, stream>>>(
        input, codes, codebooks, out, n_chunks, n_tokens);
}